// SpaceTimeLookTable_12713103196178
// MI455X (gfx1250) — compile-verified
//
#include <hip/hip_runtime.h>
#include <math.h>

// ---------------------------------------------------------------------------
// SpaceTime lookup-table NeRF-style forward, fused for MI455X (gfx1250).
//
// Exact algebraic reductions used (bit-for-bit math class, fp32 throughout):
//  * spatial_offsets are three IDENTICAL vectors -> pre-sum w1's 3 column
//    blocks per spatial table: K 112512 -> 37632, one gather row per table.
//  * rms_scale folded into w1 columns (prep kernel).
//  * per-row rmsnorm scalar applied after the GEMM.
//  * sum-of-squares: spatial blocks weighted x3, temporal x1.
//
// GEMM (M=2048 rays, N=64, K=37632) runs on V_WMMA_F32_16X16X4_F32 so the
// accumulation stays fp32 like the reference.
//
// Workspace layout: [0 .. 9,633,792) w1r (fp32, K-major [37632][64])
//                   [9,633,792 .. +64KB) per-ray gather bases (int32 [2048][8])
// ---------------------------------------------------------------------------

typedef __attribute__((ext_vector_type(2))) float v2f;
typedef __attribute__((ext_vector_type(8))) float v8f;

#define N_RAYS   2048
#define TOTAL_F  112512
#define KR       37632          // reduced K
#define NCHUNK   588            // KR / 64
#define W1R_BYTES ((size_t)KR * 64 * 4)

// ---------------------------------------------------------------------------
// Kernel 1: w1r[k][j] = sum over replicated blocks of rms_scale[c]*w1[j][c]
// K layout: [0,64) t0 | [64,576) t1 | [576,4672) t2 | [4672,37440) t3 |
//           [37440,37632) temporal (3 x 64, kept separate)
// ---------------------------------------------------------------------------
__global__ __launch_bounds__(256) void prep_w1r_kernel(
    const float* __restrict__ w1, const float* __restrict__ scale,
    float* __restrict__ w1r)
{
    int idx = blockIdx.x * 256 + threadIdx.x;
    if (idx >= KR * 64) return;
    int k = idx >> 6;
    int j = idx & 63;
    const float* wrow = w1 + (size_t)j * TOTAL_F;
    float s;
    if (k < 64) {
        int c = k;
        s = wrow[c] * scale[c] + wrow[c + 64] * scale[c + 64] + wrow[c + 128] * scale[c + 128];
    } else if (k < 576) {
        int c = 192 + (k - 64);
        s = wrow[c] * scale[c] + wrow[c + 512] * scale[c + 512] + wrow[c + 1024] * scale[c + 1024];
    } else if (k < 4672) {
        int c = 1728 + (k - 576);
        s = wrow[c] * scale[c] + wrow[c + 4096] * scale[c + 4096] + wrow[c + 8192] * scale[c + 8192];
    } else if (k < 37440) {
        int c = 14016 + (k - 4672);
        s = wrow[c] * scale[c] + wrow[c + 32768] * scale[c + 32768] + wrow[c + 65536] * scale[c + 65536];
    } else {
        int c = 112320 + (k - 37440);   // temporal passthrough
        s = wrow[c] * scale[c];
    }
    w1r[idx] = s;
}

// ---------------------------------------------------------------------------
// Kernel 2: per-ray gather base offsets (element units, all multiples of 64).
// slot: 0..3 spatial tables, 4..6 temporal rows (t-1, t, t+1). slot 7 unused.
// ---------------------------------------------------------------------------
__global__ __launch_bounds__(256) void prep_bases_kernel(
    const float* __restrict__ pos, const float* __restrict__ tvals,
    int* __restrict__ bases)
{
    int g = blockIdx.x * 256 + threadIdx.x;
    if (g >= N_RAYS) return;
    float px = pos[g * 3 + 0], py = pos[g * 3 + 1], pz = pos[g * 3 + 2];
    float tv = tvals[g];

    int x0 = (int)(px * 127.f), y0 = (int)(py * 127.f), z0 = (int)(pz * 127.f);
    int x1 = (int)(px * 63.f),  y1 = (int)(py * 63.f),  z1 = (int)(pz * 63.f);
    int x2 = (int)(px * 31.f),  y2 = (int)(py * 31.f),  z2 = (int)(pz * 31.f);
    int x3 = (int)(px * 15.f),  y3 = (int)(py * 15.f),  z3 = (int)(pz * 15.f);
    int ti = (int)(tv * 127.f);

    // spatial offset vector is (-1, 0, +1) on (x, y, z), wrapped per-axis
    int b0 = ((((x0 + 127) & 127) * 128 + y0) * 128 + ((z0 + 1) & 127)) * 64;
    int b1 = ((((x1 + 63) & 63) * 64 + y1) * 64 + ((z1 + 1) & 63)) * 512;
    int b2 = ((((x2 + 31) & 31) * 32 + y2) * 32 + ((z2 + 1) & 31)) * 4096;
    int b3 = ((((x3 + 15) & 15) * 16 + y3) * 16 + ((z3 + 1) & 15)) * 32768;
    // temporal: spatial part un-offset, t axis wrapped mod 64
    int cell = ((x3 * 16 + y3) * 16 + z3) * 64;
    int bt0 = (cell + ((ti + 63) & 63)) * 64;
    int bt1 = (cell + (ti & 63)) * 64;
    int bt2 = (cell + ((ti + 1) & 63)) * 64;

    int* o = bases + g * 8;
    o[0] = b0; o[1] = b1; o[2] = b2; o[3] = b3;
    o[4] = bt0; o[5] = bt1; o[6] = bt2; o[7] = 0;
}

// ---------------------------------------------------------------------------
// Kernel 3: fused gather + WMMA fp32 GEMM + rmsnorm + output layer.
// 64 blocks x 256 threads. Block = 32 rays x 64 inner; 8 waves, each wave
// owns one 16x16 WMMA tile (2 M-tiles x 4 N-tiles).
// ---------------------------------------------------------------------------
#define ASTRIDE 68   // padded LDS row strides (floats) to avoid bank conflicts

__global__ __launch_bounds__(256) void spacetime_fused_kernel(
    const float* __restrict__ t0, const float* __restrict__ t1,
    const float* __restrict__ t2, const float* __restrict__ t3,
    const float* __restrict__ tt,
    const float* __restrict__ w1r, const int* __restrict__ bases,
    const float* __restrict__ dirv, const float* __restrict__ tvals,
    const float* __restrict__ w2, const float* __restrict__ b2v,
    float* __restrict__ out)
{
    __shared__ float A[32 * ASTRIDE];   // gathered relu'd features, m-major
    __shared__ float B[64 * ASTRIDE];   // w1r chunk, n-major ([n][k])
    __shared__ float H[32 * ASTRIDE];   // hidden layer output
    __shared__ float SS[32];            // per-ray sum of squares
    __shared__ float INV[32];           // 1/(rms+eps)
    __shared__ int   BASE[32 * 8];

    const int tid  = threadIdx.x;
    const int ray0 = blockIdx.x * 32;

    BASE[tid] = bases[ray0 * 8 + tid];  // 256 ints exactly
    __syncthreads();

    // A-gather mapping: thread handles (ray r0, float4 q) and (ray r0+16, q)
    const int r0 = tid >> 4;        // 0..15
    const int q  = tid & 15;        // float4 index within the 64-float chunk

    // WMMA tile mapping
    const int lane  = tid & 31;
    const int wv    = tid >> 5;            // 0..7
    const int mb    = (wv >> 2) * 16;      // 0 or 16
    const int nb    = (wv & 3) * 16;       // 0,16,32,48
    const int arow  = mb + (lane & 15);
    const int brow  = nb + (lane & 15);
    const int khalf = (lane >> 4) * 2;     // A/B frag: VGPR0 holds K=0|2, VGPR1 K=1|3

    v8f acc = {};
    float ss0 = 0.f, ss1 = 0.f;

    for (int c = 0; c < NCHUNK; ++c) {
        // chunk -> (table, base slot, offset within row, sumsq weight); uniform
        const float* tab; int slot, soff; float wgt;
        if (c == 0)        { tab = t0; slot = 0; soff = 0;            wgt = 3.f; }
        else if (c < 9)    { tab = t1; slot = 1; soff = (c - 1) * 64; wgt = 3.f; }
        else if (c < 73)   { tab = t2; slot = 2; soff = (c - 9) * 64; wgt = 3.f; }
        else if (c < 585)  { tab = t3; slot = 3; soff = (c - 73) * 64; wgt = 3.f; }
        else               { tab = tt; slot = 4 + (c - 585); soff = 0; wgt = 1.f; }

        // ---- gather A (relu + weighted sumsq on the fly) ----
        {
            const float4 v = ((const float4*)(tab + BASE[r0 * 8 + slot] + soff))[q];
            float a0 = fmaxf(v.x, 0.f), a1 = fmaxf(v.y, 0.f);
            float a2 = fmaxf(v.z, 0.f), a3 = fmaxf(v.w, 0.f);
            float* ap = A + r0 * ASTRIDE + q * 4;
            ap[0] = a0; ap[1] = a1; ap[2] = a2; ap[3] = a3;
            ss0 += wgt * (a0 * a0 + a1 * a1 + a2 * a2 + a3 * a3);

            const int r1 = r0 + 16;
            const float4 u = ((const float4*)(tab + BASE[r1 * 8 + slot] + soff))[q];
            float c0 = fmaxf(u.x, 0.f), c1 = fmaxf(u.y, 0.f);
            float c2 = fmaxf(u.z, 0.f), c3 = fmaxf(u.w, 0.f);
            float* bp = A + r1 * ASTRIDE + q * 4;
            bp[0] = c0; bp[1] = c1; bp[2] = c2; bp[3] = c3;
            ss1 += wgt * (c0 * c0 + c1 * c1 + c2 * c2 + c3 * c3);
        }

        // ---- load B chunk (64 k x 64 n, k-major in global) -> n-major LDS ----
        {
            const float4* wp = (const float4*)(w1r + (size_t)c * 4096);
#pragma unroll
            for (int i = 0; i < 4; ++i) {
                int e  = tid + i * 256;      // float4 element 0..1023
                int k  = e >> 4;             // 0..63
                int n4 = (e & 15) * 4;       // inner-unit base
                float4 v = wp[e];
                B[(n4 + 0) * ASTRIDE + k] = v.x;
                B[(n4 + 1) * ASTRIDE + k] = v.y;
                B[(n4 + 2) * ASTRIDE + k] = v.z;
                B[(n4 + 3) * ASTRIDE + k] = v.w;
            }
        }
        __syncthreads();

        // ---- 16 x V_WMMA_F32_16X16X4_F32 over the 64-wide K chunk ----
#pragma unroll
        for (int kk = 0; kk < 64; kk += 4) {
            v2f a = *(const v2f*)(&A[arow * ASTRIDE + kk + khalf]);
            v2f b = *(const v2f*)(&B[brow * ASTRIDE + kk + khalf]);
            acc = __builtin_amdgcn_wmma_f32_16x16x4_f32(
                false, a, false, b, (short)0, acc, false, false);
        }
        __syncthreads();
    }

    // ---- per-ray sumsq reduction (16-thread groups live in half-waves) ----
#pragma unroll
    for (int o = 8; o >= 1; o >>= 1) {
        ss0 += __shfl_xor(ss0, o);
        ss1 += __shfl_xor(ss1, o);
    }
    if ((tid & 15) == 0) { SS[r0] = ss0; SS[r0 + 16] = ss1; }
    __syncthreads();
    if (tid < 32) {
        float rms = sqrtf(SS[tid] * (1.0f / (float)TOTAL_F));
        INV[tid] = 1.0f / (rms + 1e-8f);
    }
    __syncthreads();

    // ---- apply rmsnorm scalar, stash h tile (C layout: VGPR v -> M=v+8*(l>>4)) ----
#pragma unroll
    for (int v = 0; v < 8; ++v) {
        int row = mb + v + (lane >> 4) * 8;
        int col = nb + (lane & 15);
        H[row * ASTRIDE + col] = acc[v] * INV[row];
    }
    __syncthreads();

    // ---- output layer: out = [h, dir, t] @ w2.T + b2 ----
    if (tid < 128) {
        int r = tid >> 2, o = tid & 3;
        const float* w2o = w2 + o * 68;
        float s = b2v[o];
        const float* hr = H + r * ASTRIDE;
#pragma unroll 8
        for (int j = 0; j < 64; ++j) s += hr[j] * w2o[j];
        int g = ray0 + r;
        s += dirv[g * 3 + 0] * w2o[64] + dirv[g * 3 + 1] * w2o[65]
           + dirv[g * 3 + 2] * w2o[66] + tvals[g] * w2o[67];
        out[g * 4 + o] = s;
    }
}

// ---------------------------------------------------------------------------
extern "C" void kernel_launch(void* const* d_in, const int* in_sizes, int n_in,
                              void* d_out, int out_size, void* d_ws, size_t ws_size,
                              hipStream_t stream) {
    const float* pos    = (const float*)d_in[0];
    const float* dirv   = (const float*)d_in[1];
    const float* tvals  = (const float*)d_in[2];
    const float* table0 = (const float*)d_in[3];
    const float* table1 = (const float*)d_in[4];
    const float* table2 = (const float*)d_in[5];
    const float* table3 = (const float*)d_in[6];
    const float* ttable = (const float*)d_in[7];
    const float* scale  = (const float*)d_in[8];
    const float* w1     = (const float*)d_in[9];
    const float* w2     = (const float*)d_in[10];
    const float* b2     = (const float*)d_in[11];
    float* out = (float*)d_out;

    float* w1r  = (float*)d_ws;                                // 9,633,792 B
    int*   bases = (int*)((char*)d_ws + W1R_BYTES);            // + 64 KB

    prep_w1r_kernel<<<(KR * 64 + 255) / 256, 256, 0, stream>>>(w1, scale, w1r);
    prep_bases_kernel<<<(N_RAYS + 255) / 256, 256, 0, stream>>>(pos, tvals, bases);
    spacetime_fused_kernel<<<64, 256, 0, stream>>>(
        table0, table1, table2, table3, ttable, w1r, bases,
        dirv, tvals, w2, b2, out);
}